// GlobalPool_90726889161262
// MI455X (gfx1250) — compile-verified
//
#include <hip/hip_runtime.h>
#include <math.h>

typedef __attribute__((ext_vector_type(2))) float v2f;
typedef __attribute__((ext_vector_type(8))) float v8f;

#define FEAT 256

// ---------------------------------------------------------------------------
// Kernel 1: per-graph attention pooling (fused logit + softmax + weighted sum)
// one workgroup (256 threads = 8 wave32) per graph
// ---------------------------------------------------------------------------
__global__ void pool_kernel(const float* __restrict__ node_feats,
                            const float* __restrict__ g_feats,
                            const int*   __restrict__ seg,
                            const float* __restrict__ logit_w,
                            const float* __restrict__ logit_b,
                            int n_nodes,
                            float* __restrict__ z_buf,
                            float* __restrict__ pooled,
                            float* __restrict__ s_buf) {
    const int g    = blockIdx.x;
    const int t    = threadIdx.x;
    const int lane = t & 31;
    const int wave = t >> 5;

    __shared__ float sw[FEAT];        // node-half of logit weights
    __shared__ float sred[8];
    __shared__ float smax[8];
    __shared__ float sS[8];
    __shared__ float sacc[8][FEAT];
    __shared__ float sC;
    __shared__ float sM;

    // --- segment bounds via binary search (seg is sorted) ---
    int lo, hi;
    {
        int a = 0, b = n_nodes;
        while (a < b) { int mid = (a + b) >> 1; if (seg[mid] < g) a = mid + 1; else b = mid; }
        lo = a;
        a = lo; b = n_nodes;
        while (a < b) { int mid = (a + b) >> 1; if (seg[mid] < g + 1) a = mid + 1; else b = mid; }
        hi = a;
    }

    // --- c = relu(g_feats[g]) . logit_w[0:256] + b  (segment-constant part) ---
    {
        float gv = g_feats[(size_t)g * FEAT + t];
        gv = gv > 0.0f ? gv : 0.0f;
        float p = gv * logit_w[t];
        #pragma unroll
        for (int off = 16; off > 0; off >>= 1) p += __shfl_xor(p, off, 32);
        if (lane == 0) sred[wave] = p;
    }
    sw[t] = logit_w[FEAT + t];
    __syncthreads();
    if (t == 0) {
        float sum = 0.0f;
        #pragma unroll
        for (int w = 0; w < 8; ++w) sum += sred[w];
        sC = sum + logit_b[0];
    }
    __syncthreads();
    const float cbase = sC;

    // --- pass 1: z_n = leaky_relu(c + feat_n . w_node), track max ---
    float mloc = -3.4e38f;
    const int kb = lane * 8;
    for (int n = lo + wave; n < hi; n += 8) {
        const float* row = node_feats + (size_t)n * FEAT + kb;
        float4 f0 = *(const float4*)(row);
        float4 f1 = *(const float4*)(row + 4);
        float d = f0.x * sw[kb + 0] + f0.y * sw[kb + 1] + f0.z * sw[kb + 2] + f0.w * sw[kb + 3]
                + f1.x * sw[kb + 4] + f1.y * sw[kb + 5] + f1.z * sw[kb + 6] + f1.w * sw[kb + 7];
        #pragma unroll
        for (int off = 16; off > 0; off >>= 1) d += __shfl_xor(d, off, 32);
        float zz = cbase + d;
        zz = zz > 0.0f ? zz : 0.01f * zz;   // leaky_relu
        if (lane == 0) z_buf[n] = zz;
        mloc = fmaxf(mloc, zz);             // uniform across the wave
    }
    if (lane == 0) smax[wave] = mloc;
    __syncthreads();
    if (t == 0) {
        float m = -3.4e38f;
        #pragma unroll
        for (int w = 0; w < 8; ++w) m = fmaxf(m, smax[w]);
        sM = m;
    }
    __syncthreads();
    const float m = sM;

    // --- pass 2: acc = sum e_n * feat_n ; s = sum e_n (simultaneously) ---
    float accv[8];
    #pragma unroll
    for (int j = 0; j < 8; ++j) accv[j] = 0.0f;
    float sloc = 0.0f;
    for (int n = lo + wave; n < hi; n += 8) {
        float e = __expf(z_buf[n] - m);
        const float* row = node_feats + (size_t)n * FEAT + kb;
        __builtin_prefetch(row + 8 * FEAT, 0, 0);   // global_prefetch_b8
        float4 f0 = *(const float4*)(row);
        float4 f1 = *(const float4*)(row + 4);
        accv[0] += e * f0.x; accv[1] += e * f0.y; accv[2] += e * f0.z; accv[3] += e * f0.w;
        accv[4] += e * f1.x; accv[5] += e * f1.y; accv[6] += e * f1.z; accv[7] += e * f1.w;
        sloc += e;
    }
    #pragma unroll
    for (int j = 0; j < 8; ++j) sacc[wave][kb + j] = accv[j];
    if (lane == 0) sS[wave] = sloc;
    __syncthreads();

    float tot = 0.0f, s = 0.0f;
    #pragma unroll
    for (int w = 0; w < 8; ++w) { tot += sacc[w][t]; s += sS[w]; }
    pooled[(size_t)g * FEAT + t] = (s > 0.0f) ? (tot / s) : 0.0f;
    if (t == 0) s_buf[g] = s;
}

// ---------------------------------------------------------------------------
// Kernel 2a: X = elu(pooled @ proj_w.T + proj_b)   [1024x256, K=256]
// one wave32 per 16x16 tile, V_WMMA_F32_16X16X4_F32
// ---------------------------------------------------------------------------
__global__ void proj_elu_wmma(const float* __restrict__ pooled,
                              const float* __restrict__ proj_w,
                              const float* __restrict__ proj_b,
                              const float* __restrict__ s_buf,
                              float* __restrict__ X) {
    const int m0   = blockIdx.x * 16;
    const int n0   = blockIdx.y * 16;
    const int lane = threadIdx.x;
    const int l15  = lane & 15;
    const int hl   = lane >> 4;      // which K-pair half this lane holds

    v8f c = {};
    const float* arow = pooled + (size_t)(m0 + l15) * FEAT + 2 * hl;
    const float* brow = proj_w + (size_t)(n0 + l15) * FEAT + 2 * hl;
    for (int k = 0; k < FEAT; k += 4) {
        v2f a = *(const v2f*)(arow + k);
        v2f b = *(const v2f*)(brow + k);
        c = __builtin_amdgcn_wmma_f32_16x16x4_f32(false, a, false, b, (short)0, c,
                                                  false, false);
    }
    #pragma unroll
    for (int r = 0; r < 8; ++r) {
        int row = m0 + r + 8 * hl;
        int col = n0 + l15;
        float bias = (s_buf[row] > 0.0f) ? proj_b[col] : 0.0f;  // empty-segment guard
        float v = c[r] + bias;
        v = v > 0.0f ? v : (__expf(v) - 1.0f);                  // elu
        X[(size_t)row * FEAT + col] = v;
    }
}

// ---------------------------------------------------------------------------
// Kernel 2b: fused GRU cell — 6 WMMA tile-accumulators (3 gates x {x,h})
// gi = X @ w_ih.T + b_ih ; gh = g_feats @ w_hh.T + b_hh ; combine
// ---------------------------------------------------------------------------
__global__ void gru_wmma(const float* __restrict__ X,
                         const float* __restrict__ g_feats,
                         const float* __restrict__ w_ih,
                         const float* __restrict__ w_hh,
                         const float* __restrict__ b_ih,
                         const float* __restrict__ b_hh,
                         float* __restrict__ out) {
    const int m0   = blockIdx.x * 16;   // graph rows
    const int f0   = blockIdx.y * 16;   // feature cols (0..255)
    const int lane = threadIdx.x;
    const int l15  = lane & 15;
    const int hl   = lane >> 4;

    v8f ci0 = {}, ci1 = {}, ci2 = {};   // gi gates r,z,n
    v8f ch0 = {}, ch1 = {}, ch2 = {};   // gh gates r,z,n

    const float* xrow = X       + (size_t)(m0 + l15) * FEAT + 2 * hl;
    const float* hrow = g_feats + (size_t)(m0 + l15) * FEAT + 2 * hl;
    const float* bi0  = w_ih + (size_t)(      f0 + l15) * FEAT + 2 * hl;
    const float* bi1  = w_ih + (size_t)(256 + f0 + l15) * FEAT + 2 * hl;
    const float* bi2  = w_ih + (size_t)(512 + f0 + l15) * FEAT + 2 * hl;
    const float* bh0  = w_hh + (size_t)(      f0 + l15) * FEAT + 2 * hl;
    const float* bh1  = w_hh + (size_t)(256 + f0 + l15) * FEAT + 2 * hl;
    const float* bh2  = w_hh + (size_t)(512 + f0 + l15) * FEAT + 2 * hl;

    for (int k = 0; k < FEAT; k += 4) {
        v2f ax = *(const v2f*)(xrow + k);
        v2f ah = *(const v2f*)(hrow + k);
        v2f w0 = *(const v2f*)(bi0 + k);
        v2f w1 = *(const v2f*)(bi1 + k);
        v2f w2 = *(const v2f*)(bi2 + k);
        v2f u0 = *(const v2f*)(bh0 + k);
        v2f u1 = *(const v2f*)(bh1 + k);
        v2f u2 = *(const v2f*)(bh2 + k);
        ci0 = __builtin_amdgcn_wmma_f32_16x16x4_f32(false, ax, false, w0, (short)0, ci0, false, false);
        ci1 = __builtin_amdgcn_wmma_f32_16x16x4_f32(false, ax, false, w1, (short)0, ci1, false, false);
        ci2 = __builtin_amdgcn_wmma_f32_16x16x4_f32(false, ax, false, w2, (short)0, ci2, false, false);
        ch0 = __builtin_amdgcn_wmma_f32_16x16x4_f32(false, ah, false, u0, (short)0, ch0, false, false);
        ch1 = __builtin_amdgcn_wmma_f32_16x16x4_f32(false, ah, false, u1, (short)0, ch1, false, false);
        ch2 = __builtin_amdgcn_wmma_f32_16x16x4_f32(false, ah, false, u2, (short)0, ch2, false, false);
    }

    #pragma unroll
    for (int r = 0; r < 8; ++r) {
        int row = m0 + r + 8 * hl;
        int col = f0 + l15;
        float ir = ci0[r] + b_ih[col];
        float iz = ci1[r] + b_ih[col + 256];
        float in_ = ci2[r] + b_ih[col + 512];
        float hr = ch0[r] + b_hh[col];
        float hz = ch1[r] + b_hh[col + 256];
        float hn = ch2[r] + b_hh[col + 512];
        float rg = 1.0f / (1.0f + __expf(-(ir + hr)));
        float zg = 1.0f / (1.0f + __expf(-(iz + hz)));
        float ng = tanhf(in_ + rg * hn);
        float h  = g_feats[(size_t)row * FEAT + col];
        out[(size_t)row * FEAT + col] = (1.0f - zg) * ng + zg * h;
    }
}

// ---------------------------------------------------------------------------
extern "C" void kernel_launch(void* const* d_in, const int* in_sizes, int n_in,
                              void* d_out, int out_size, void* d_ws, size_t ws_size,
                              hipStream_t stream) {
    const float* node_feats = (const float*)d_in[0];
    const float* g_feats    = (const float*)d_in[1];
    const int*   seg        = (const int*)  d_in[2];
    // d_in[3] = num_graphs (device scalar; dims derived host-side instead)
    const float* logit_w    = (const float*)d_in[4];
    const float* logit_b    = (const float*)d_in[5];
    const float* proj_w     = (const float*)d_in[6];
    const float* proj_b     = (const float*)d_in[7];
    const float* w_ih       = (const float*)d_in[8];
    const float* w_hh       = (const float*)d_in[9];
    const float* b_ih       = (const float*)d_in[10];
    const float* b_hh       = (const float*)d_in[11];
    float*       out        = (float*)d_out;

    const int n_nodes  = in_sizes[0] / FEAT;
    const int n_graphs = in_sizes[1] / FEAT;

    float* ws     = (float*)d_ws;
    float* z_buf  = ws;                                  // n_nodes
    float* pooled = z_buf + n_nodes;                     // n_graphs*FEAT
    float* s_buf  = pooled + (size_t)n_graphs * FEAT;    // n_graphs
    float* Xbuf   = s_buf + ((n_graphs + 63) / 64) * 64; // n_graphs*FEAT

    pool_kernel<<<n_graphs, 256, 0, stream>>>(node_feats, g_feats, seg, logit_w,
                                              logit_b, n_nodes, z_buf, pooled, s_buf);

    dim3 gridA(n_graphs / 16, FEAT / 16);
    proj_elu_wmma<<<gridA, 32, 0, stream>>>(pooled, proj_w, proj_b, s_buf, Xbuf);

    dim3 gridB(n_graphs / 16, FEAT / 16);
    gru_wmma<<<gridB, 32, 0, stream>>>(Xbuf, g_feats, w_ih, w_hh, b_ih, b_hh, out);
}